// PVABlock_1580547974594
// MI455X (gfx1250) — compile-verified
//
#include <hip/hip_runtime.h>

// ---------------------------------------------------------------------------
// PVA attention block for MI455X (gfx1250, wave32, WMMA).
//   B=8, C=IN=INNER=EMB=512, HW=1024, S=32
//   k1: o_out[32,512]   = OE @ Wo^T                (WMMA f32 16x16x4)
//   k2: per (b, n-panel of 16): v = xf @ Wv^T via WMMA -> LDS,
//       att_raw[b,s,n] = sum_i tanh(o_out[s,i]+v[n,i])*We[i]   (fused, no t tensor)
//   k3: softmax over n + out[b,s,c] = sum_n att[n]*x[b,c,n]
// ---------------------------------------------------------------------------

typedef __attribute__((ext_vector_type(2))) float v2f;
typedef __attribute__((ext_vector_type(8))) float v8f;

#define NSTEPS 32
#define CH     512
#define HW     1024
#define NB     8
#define VROW   520   // padded LDS row stride for the v panel (16 x 520 floats)

__device__ __forceinline__ float tanh_fast(float x) {
    // tanh(x) = 1 - 2/(exp(2x)+1); exact limits at +-inf, runs on v_exp_f32
    float e = __expf(2.0f * x);
    return 1.0f - 2.0f / (e + 1.0f);
}

// ---------------------------------------------------------------------------
// Kernel 1: o_out = OE @ Wo^T   (M=32, N=512, K=512), 1 block x 8 waves.
// A[m,k]=OE[s0+m,k]; B[k,j]=Wo[i0+j,k]. f32 WMMA 16x16x4 lane layout:
//   lanes 0-15: K pair {kb,kb+1}; lanes 16-31: {kb+2,kb+3}; m/j = lane&15.
// ---------------------------------------------------------------------------
__global__ __launch_bounds__(256) void pva_oout(const float* __restrict__ OE,
                                                const float* __restrict__ Wo,
                                                float* __restrict__ o_out) {
    const int tid  = threadIdx.x;
    const int w    = tid >> 5;
    const int lane = tid & 31;
    const int m    = lane & 15;
    const int kh   = lane >> 4;          // 0 or 1 -> K offset 0 or 2

    for (int t = w; t < 64; t += 8) {
        const int s0 = (t >> 5) << 4;    // 2 M-tiles
        const int i0 = (t & 31) << 4;    // 32 N-tiles
        const float* arow = OE + (size_t)(s0 + m) * CH + 2 * kh;
        const float* brow = Wo + (size_t)(i0 + m) * CH + 2 * kh;
        v8f c = {};
        for (int kb = 0; kb < CH; kb += 4) {
            v2f a  = *(const v2f*)(arow + kb);
            v2f bv = *(const v2f*)(brow + kb);
            c = __builtin_amdgcn_wmma_f32_16x16x4_f32(false, a, false, bv,
                                                      (short)0, c, false, false);
        }
#pragma unroll
        for (int v = 0; v < 8; ++v)
            o_out[(size_t)(s0 + v + (kh << 3)) * CH + i0 + m] = c[v];
    }
}

// ---------------------------------------------------------------------------
// Kernel 2: fused v-panel GEMM + tanh attention scores.
// grid = 8 batches * 64 n-panels; block = 256 threads (8 waves).
// ---------------------------------------------------------------------------
__global__ __launch_bounds__(256) void pva_vout_att(const float* __restrict__ x,
                                                    const float* __restrict__ Wv,
                                                    const float* __restrict__ We,
                                                    const float* __restrict__ o_out,
                                                    float* __restrict__ att) {
    __shared__ float lds[16 * VROW];     // 33280 B; reused: A panel then v panel
    const int tid = threadIdx.x;
    const int b   = blockIdx.x >> 6;
    const int n0  = (blockIdx.x & 63) << 4;

    // ---- Phase A: stage A panel: lds[k*16+m] = x[b, k, n0+m]  (k=c, m=n) ----
    {
        const int r0 = tid >> 2;
        const int q  = (tid & 3) << 2;
        const float* xb = x + (size_t)b * CH * HW + n0 + q;
        for (int it = 0; it < 8; ++it) {
            const int r = r0 + (it << 6);
            float4 v = *(const float4*)(xb + (size_t)r * HW);
            *(float4*)(&lds[r * 16 + q]) = v;
        }
    }
    __syncthreads();

    // ---- Phase B: v[16 x 512] = A @ Wv^T, accumulators stay in VGPRs ----
    const int lane = tid & 31;
    const int w    = tid >> 5;
    const int m    = lane & 15;
    const int kh   = lane >> 4;
    v8f acc[4] = {};
    for (int p = 0; p < 4; ++p) {
        const int i0 = (p << 7) + (w << 4);
        const float* wrow = Wv + (size_t)(i0 + m) * CH + 2 * kh;  // B[k,j]=Wv[i0+j,k]
        v8f c = acc[p];
        for (int kb = 0; kb < CH; kb += 4) {
            v2f a;
            a.x = lds[(kb + 2 * kh)     * 16 + m];
            a.y = lds[(kb + 2 * kh + 1) * 16 + m];
            v2f bv = *(const v2f*)(wrow + kb);
            c = __builtin_amdgcn_wmma_f32_16x16x4_f32(false, a, false, bv,
                                                      (short)0, c, false, false);
        }
        acc[p] = c;
    }
    __syncthreads();

    // ---- Phase C: spill v panel to LDS: lds[n*VROW + i] ----
    for (int p = 0; p < 4; ++p) {
        const int i0 = (p << 7) + (w << 4);
#pragma unroll
        for (int v = 0; v < 8; ++v) {
            const int row = v + (kh << 3);          // n within panel
            lds[row * VROW + i0 + m] = acc[p][v];   // col j == m
        }
    }
    __syncthreads();

    // ---- Phase D: att[b,s,n0+n] = sum_i tanh(o_out[s,i]+v[n,i])*We[i] ----
    {
        const int s  = tid >> 3;                    // 32 steps
        const int nl = tid & 7;                     // 2 n-rows per thread
        const float* orow = o_out + (size_t)s * CH;
        for (int half = 0; half < 2; ++half) {
            const int n = nl + (half << 3);
            float sum = 0.f;
            for (int i = 0; i < CH; i += 4) {
                float4 ov = *(const float4*)(orow + i);
                float4 vv = *(const float4*)(&lds[n * VROW + i]);
                float4 we = *(const float4*)(We + i);
                sum += tanh_fast(ov.x + vv.x) * we.x;
                sum += tanh_fast(ov.y + vv.y) * we.y;
                sum += tanh_fast(ov.z + vv.z) * we.z;
                sum += tanh_fast(ov.w + vv.w) * we.w;
            }
            att[(size_t)(b * NSTEPS + s) * HW + n0 + n] = sum;
        }
    }
}

// ---------------------------------------------------------------------------
// Kernel 3: softmax over hw + out[b,s,c] = sum_n att[n] * x[b,c,n]
// grid = 8*32 blocks; block = 256 threads; x rows are contiguous in n.
// ---------------------------------------------------------------------------
__global__ __launch_bounds__(256) void pva_softmax_out(const float* __restrict__ x,
                                                       const float* __restrict__ att,
                                                       float* __restrict__ out) {
    __shared__ float wsh[HW];
    __shared__ float red[256];
    const int tid = threadIdx.x;
    const int b   = blockIdx.x >> 5;
    const int s   = blockIdx.x & 31;
    const float* arow = att + (size_t)(b * NSTEPS + s) * HW;

    float m = -3.402823466e38f;
    for (int k = tid; k < HW; k += 256) m = fmaxf(m, arow[k]);
    red[tid] = m;
    __syncthreads();
    for (int off = 128; off > 0; off >>= 1) {
        if (tid < off) red[tid] = fmaxf(red[tid], red[tid + off]);
        __syncthreads();
    }
    m = red[0];
    __syncthreads();

    float sum = 0.f;
    for (int k = tid; k < HW; k += 256) {
        float e = __expf(arow[k] - m);
        wsh[k] = e;
        sum += e;
    }
    red[tid] = sum;
    __syncthreads();
    for (int off = 128; off > 0; off >>= 1) {
        if (tid < off) red[tid] += red[tid + off];
        __syncthreads();
    }
    const float inv = 1.0f / red[0];

    for (int ci = 0; ci < 2; ++ci) {
        const int c = tid + (ci << 8);
        const float* xrow = x + ((size_t)b * CH + c) * HW;  // contiguous in n
        float acc = 0.f;
        for (int k = 0; k < HW; k += 4) {
            float4 xv = *(const float4*)(xrow + k);
            float4 wv = *(const float4*)(&wsh[k]);
            acc = fmaf(xv.x, wv.x, acc);
            acc = fmaf(xv.y, wv.y, acc);
            acc = fmaf(xv.z, wv.z, acc);
            acc = fmaf(xv.w, wv.w, acc);
        }
        out[(size_t)(b * NSTEPS + s) * CH + c] = acc * inv;
    }
}

// ---------------------------------------------------------------------------
extern "C" void kernel_launch(void* const* d_in, const int* in_sizes, int n_in,
                              void* d_out, int out_size, void* d_ws, size_t ws_size,
                              hipStream_t stream) {
    const float* x  = (const float*)d_in[0];   // [8,512,16,64]
    const float* OE = (const float*)d_in[1];   // [32,512]
    const float* Wv = (const float*)d_in[2];   // [512,512]
    const float* Wo = (const float*)d_in[3];   // [512,512]
    const float* We = (const float*)d_in[4];   // [1,512]
    float* out = (float*)d_out;                // [8,32,512]

    float* wsf   = (float*)d_ws;
    float* o_out = wsf;                        // 32*512 floats   (64 KB)
    float* att   = wsf + NSTEPS * CH;          // 8*32*1024 floats (1 MB)

    pva_oout<<<1, 256, 0, stream>>>(OE, Wo, o_out);
    pva_vout_att<<<NB * (HW / 16), 256, 0, stream>>>(x, Wv, We, o_out, att);
    pva_softmax_out<<<NB * NSTEPS, 256, 0, stream>>>(x, att, out);
}